// Net_32650341384626
// MI455X (gfx1250) — compile-verified
//
#include <hip/hip_runtime.h>
#include <hip/hip_bf16.h>
#include <math.h>

typedef __attribute__((ext_vector_type(2))) float v2f;
typedef __attribute__((ext_vector_type(8))) float v8f;

#define THREADS 256
#define WAVES_PER_BLK (THREADS / 32)

// ---------------- degree + index narrowing ----------------

__global__ void k_init_deg(float* __restrict__ deg, int n) {
  int i = blockIdx.x * blockDim.x + threadIdx.x;
  if (i < n) deg[i] = 1.0f;  // self-loop contributes 1
}

// Single streaming pass over the int64 edge list: narrow to int32 (all later
// passes read half the index bytes) and accumulate in-degree.
__global__ void k_cvt_deg(const long long* __restrict__ src64,
                          const long long* __restrict__ dst64,
                          int* __restrict__ src32,
                          int* __restrict__ dst32,
                          float* __restrict__ deg, int e) {
  int i = blockIdx.x * blockDim.x + threadIdx.x;
  if (i < e) {
    int s = (int)src64[i];
    int d = (int)dst64[i];
    src32[i] = s;
    dst32[i] = d;
    atomicAdd(&deg[d], 1.0f);
  }
}

// dinv[i] = 1/sqrt(deg); self-loop init of the 3-wide aggregate: agg1 = x * dinv^2
__global__ void k_dinv_self3(const float* __restrict__ deg,
                             const float* __restrict__ x,
                             float* __restrict__ dinv,
                             float* __restrict__ agg1, int n) {
  int i = blockIdx.x * blockDim.x + threadIdx.x;
  if (i < n) {
    float di = 1.0f / sqrtf(deg[i]);
    dinv[i] = di;
    float s = di * di;
    agg1[i * 3 + 0] = x[i * 3 + 0] * s;
    agg1[i * 3 + 1] = x[i * 3 + 1] * s;
    agg1[i * 3 + 2] = x[i * 3 + 2] * s;
  }
}

// ---------------- edge propagation (scatter-add) ----------------

// 4 lanes per edge: lanes 0..2 scatter the 3 features, lane 3 (otherwise idle)
// persists the per-edge symmetric norm for the later passes.
__global__ void k_prop3(const int* __restrict__ src,
                        const int* __restrict__ dst,
                        const float* __restrict__ x,
                        const float* __restrict__ dinv,
                        float* __restrict__ agg1,
                        float* __restrict__ nrm, int e) {
  int t = blockIdx.x * blockDim.x + threadIdx.x;
  int edge = t >> 2, f = t & 3;
  if (edge < e) {
    int s = src[edge], d = dst[edge];
    float nm = dinv[s] * dinv[d];
    if (f < 3) {
      atomicAdd(&agg1[d * 3 + f], x[s * 3 + f] * nm);
    } else {
      nrm[edge] = nm;
    }
  }
}

// 16 lanes per edge: coalesced 64B gather of hw2[src], 16 f32 atomics to agg2[dst].
// dinv gathers replaced by one broadcast norm read (12B/edge of index+norm total).
__global__ void k_prop16(const int* __restrict__ src,
                         const int* __restrict__ dst,
                         const float* __restrict__ nrm,
                         const float* __restrict__ hw2,
                         float* __restrict__ agg2, int e) {
  int t = blockIdx.x * blockDim.x + threadIdx.x;
  int edge = t >> 4, f = t & 15;
  if (edge < e) {
    int s = src[edge], d = dst[edge];
    float nm = nrm[edge];
    atomicAdd(&agg2[d * 16 + f], hw2[s * 16 + f] * nm);
  }
}

__global__ void k_prop1(const int* __restrict__ src,
                        const int* __restrict__ dst,
                        const float* __restrict__ nrm,
                        const float* __restrict__ sbuf,
                        float* __restrict__ agg3, int e) {
  int i = blockIdx.x * blockDim.x + threadIdx.x;
  if (i < e) {
    atomicAdd(&agg3[dst[i]], sbuf[src[i]] * nrm[i]);
  }
}

// ---------------- fused per-node transforms via WMMA f32 16x16x4 ----------------
// Per 16-node tile (one wave):
//   T   = relu(agg1(16x3, K padded to 4) @ W_in + b_in)   -- 1 WMMA
//   hw2 = T @ W_mid                                       -- 4 WMMAs (K=16)
//   agg2 self-loop init = hw2 * dinv^2
// A layout (16x4 f32): lanes 0-15 hold K=0,1 ; lanes 16-31 hold K=2,3 (row = lane&15)
// B layout (4x16 f32): vgpr v: lanes 0-15 -> K=v, lanes 16-31 -> K=v+2, col = lane&15
// C/D layout: vgpr v: lanes 0-15 -> M=v, lanes 16-31 -> M=v+8, col = lane&15
__global__ void __launch_bounds__(THREADS)
k_feat_wmma(const float* __restrict__ agg1,  // [n,3]
            const float* __restrict__ Win,   // [3,16]
            const float* __restrict__ bin,   // [16]
            const float* __restrict__ Wmid,  // [16,16]
            const float* __restrict__ dinv,
            float* __restrict__ hw2,         // [n,16]
            float* __restrict__ agg2,        // [n,16]
            int n, int ntiles) {
  __shared__ float lds[WAVES_PER_BLK][16 * 17];  // stride-17 pad: conflict-free transpose
  const int waveInBlk = threadIdx.x >> 5;
  const int lane = threadIdx.x & 31;
  int tile = blockIdx.x * WAVES_PER_BLK + waveInBlk;
  const bool live = tile < ntiles;
  if (!live) tile = ntiles - 1;          // clamp whole wave (uniform, EXEC stays full)
  const int base = tile * 16;
  const int row = lane & 15;             // = M for A, = N(col) for B/C/D
  const int hi = lane >> 4;
  int node = base + row;
  if (node >= n) node = n - 1;           // select, not a branch
  float* tld = lds[waveInBlk];

  // ---- layer-1 transform: relu(agg1 @ W_in + b_in) ----
  {
    const int k0 = hi * 2;               // 0 or 2; K=3 slot is zero-padded
    v2f a, b;
    a.x = agg1[node * 3 + k0];
    a.y = (k0 + 1 < 3) ? agg1[node * 3 + k0 + 1] : 0.0f;
    b.x = Win[k0 * 16 + row];
    b.y = (k0 + 1 < 3) ? Win[(k0 + 1) * 16 + row] : 0.0f;
    v8f c = {0.f, 0.f, 0.f, 0.f, 0.f, 0.f, 0.f, 0.f};
    c = __builtin_amdgcn_wmma_f32_16x16x4_f32(false, a, false, b, (short)0, c,
                                              false, false);
    const float bias = bin[row];
#pragma unroll
    for (int v = 0; v < 8; ++v) {
      float h = c[v] + bias;
      h = h > 0.0f ? h : 0.0f;
      tld[(v + hi * 8) * 17 + row] = h;  // element (M = v+hi*8, col = row)
    }
  }
  __syncthreads();  // all 256 threads reach this (no early returns)

  // ---- layer-2 transform: hw2 = T @ W_mid (K=16 in 4 chunks) ----
  v8f c2 = {0.f, 0.f, 0.f, 0.f, 0.f, 0.f, 0.f, 0.f};
#pragma unroll
  for (int kk = 0; kk < 16; kk += 4) {
    const int ka = kk + hi * 2;
    v2f a2, b2;
    a2.x = tld[row * 17 + ka];
    a2.y = tld[row * 17 + ka + 1];
    b2.x = Wmid[ka * 16 + row];
    b2.y = Wmid[(ka + 1) * 16 + row];
    c2 = __builtin_amdgcn_wmma_f32_16x16x4_f32(false, a2, false, b2, (short)0, c2,
                                               false, false);
  }

  if (live) {
#pragma unroll
    for (int v = 0; v < 8; ++v) {
      const int m = base + v + hi * 8;
      if (m < n) {
        const float val = c2[v];
        hw2[m * 16 + row] = val;
        const float di = dinv[m];
        agg2[m * 16 + row] = val * di * di;  // self-loop init
      }
    }
  }
}

// ---------------- output head ----------------

// h2 = relu(agg2 + b_mid); s = h2 @ W_out; agg3 self-loop init = s * dinv^2
__global__ void k_out(const float* __restrict__ agg2,
                      const float* __restrict__ bmid,
                      const float* __restrict__ Wout,
                      const float* __restrict__ dinv,
                      float* __restrict__ sbuf,
                      float* __restrict__ agg3, int n) {
  int i = blockIdx.x * blockDim.x + threadIdx.x;
  if (i < n) {
    float acc = 0.0f;
#pragma unroll
    for (int j = 0; j < 16; ++j) {
      float h = agg2[i * 16 + j] + bmid[j];
      h = h > 0.0f ? h : 0.0f;
      acc += h * Wout[j];
    }
    sbuf[i] = acc;
    float di = dinv[i];
    agg3[i] = acc * di * di;
  }
}

__global__ void k_final(const float* __restrict__ agg3,
                        const float* __restrict__ bout,
                        float* __restrict__ out, int n) {
  int i = blockIdx.x * blockDim.x + threadIdx.x;
  if (i < n) {
    float z = agg3[i] + bout[0];
    out[i] = 1.0f / (1.0f + expf(-z));
  }
}

// ---------------- launch ----------------

extern "C" void kernel_launch(void* const* d_in, const int* in_sizes, int n_in,
                              void* d_out, int out_size, void* d_ws, size_t ws_size,
                              hipStream_t stream) {
  const float* x = (const float*)d_in[0];
  const long long* ei = (const long long*)d_in[1];  // int64 [2,E]
  const float* Win = (const float*)d_in[2];
  const float* bin = (const float*)d_in[3];
  const float* Wmid = (const float*)d_in[4];  // [1,16,16]
  const float* bmid = (const float*)d_in[5];  // [1,16]
  const float* Wout = (const float*)d_in[6];  // [16,1]
  const float* bout = (const float*)d_in[7];  // [1]

  const int n = in_sizes[0] / 3;
  const int e = in_sizes[1] / 2;
  const long long* src64 = ei;
  const long long* dst64 = ei + e;

  char* ws = (char*)d_ws;
  size_t off = 0;
  auto alloc = [&](size_t bytes) -> void* {
    void* p = (void*)(ws + off);
    off = (off + bytes + 255) & ~(size_t)255;
    return p;
  };
  float* deg   = (float*)alloc((size_t)n * 4);
  float* dinv  = (float*)alloc((size_t)n * 4);
  float* agg1  = (float*)alloc((size_t)n * 12);
  float* hw2   = (float*)alloc((size_t)n * 64);
  float* agg2  = (float*)alloc((size_t)n * 64);
  float* sbuf  = (float*)alloc((size_t)n * 4);
  float* agg3  = (float*)alloc((size_t)n * 4);
  int*   src32 = (int*)alloc((size_t)e * 4);
  int*   dst32 = (int*)alloc((size_t)e * 4);
  float* nrm   = (float*)alloc((size_t)e * 4);

  const int gN = (n + THREADS - 1) / THREADS;
  const int gE = (e + THREADS - 1) / THREADS;
  const int ntiles = (n + 15) / 16;
  const int gW = (ntiles + WAVES_PER_BLK - 1) / WAVES_PER_BLK;

  // degrees + int64 -> int32 narrowing (single pass over the big edge list)
  k_init_deg<<<gN, THREADS, 0, stream>>>(deg, n);
  k_cvt_deg<<<gE, THREADS, 0, stream>>>(src64, dst64, src32, dst32, deg, e);
  k_dinv_self3<<<gN, THREADS, 0, stream>>>(deg, x, dinv, agg1, n);

  // layer-1 aggregation over raw 3-wide features; also materializes per-edge norm
  k_prop3<<<(4 * e + THREADS - 1) / THREADS, THREADS, 0, stream>>>(
      src32, dst32, x, dinv, agg1, nrm, e);

  // fused (3->16 + relu) then (16x16) transforms on the WMMA pipe
  k_feat_wmma<<<gW, THREADS, 0, stream>>>(agg1, Win, bin, Wmid, dinv, hw2, agg2, n, ntiles);

  // layer-2 aggregation (16-wide, 16 lanes/edge, coalesced gather)
  k_prop16<<<(16 * e + THREADS - 1) / THREADS, THREADS, 0, stream>>>(
      src32, dst32, nrm, hw2, agg2, e);

  // head: relu+bias, 16->1 transform, 1-wide aggregation, sigmoid
  k_out<<<gN, THREADS, 0, stream>>>(agg2, bmid, Wout, dinv, sbuf, agg3, n);
  k_prop1<<<gE, THREADS, 0, stream>>>(src32, dst32, nrm, sbuf, agg3, e);
  k_final<<<gN, THREADS, 0, stream>>>(agg3, bout, (float*)d_out, n);
}